// BatchedDynamicThresholdLIF_5875515261285
// MI455X (gfx1250) — compile-verified
//
#include <hip/hip_runtime.h>
#include <stdint.h>

// ---------------------------------------------------------------------------
// Batched dynamic-threshold LIF scan, MI455X (gfx1250).
// Memory-bound streaming recurrence: 524 MB total traffic -> ~22.5 us floor
// at 23.3 TB/s. No matmul structure => WMMA inapplicable; the CDNA5 feature
// that matters is the ASYNCcnt-tracked global->LDS async copy engine used as
// a deep (16-step) per-wave prefetch ring to maximize memory-level
// parallelism while the VALU runs the per-neuron recurrence from registers.
// ---------------------------------------------------------------------------

#define T_STEPS   1000
#define VEC       2            // floats per thread per timestep (b64 accesses)
#define BLOCK     256          // 8 waves per block (wave32)
#define DEPTH     16           // async prefetch depth (ring slots)
#define SLOT_MASK (DEPTH - 1)

#define K_V_REST   (-65.0f)
#define K_THR0     (-50.0f)
#define K_TAUM_INV (1.0f / 20.0f)
#define K_TAUTH_INV (1.0f / 100.0f)
#define K_BETA     (5.0f)

#if defined(__has_builtin)
#  if __has_builtin(__builtin_amdgcn_global_load_async_to_lds_b64) && \
      __has_builtin(__builtin_amdgcn_s_wait_asynccnt)
#    define ASYNC_PATH 1
#  endif
#endif
#ifndef ASYNC_PATH
#  define ASYNC_PATH 0
#endif

// Payload type the async-copy builtins expect: 64-bit = <2 x i32>.
typedef int v2i __attribute__((__vector_size__(2 * sizeof(int))));
typedef v2i __attribute__((address_space(1)))* gv2i_ptr;   // global
typedef v2i __attribute__((address_space(3)))* lv2i_ptr;   // LDS

__device__ __forceinline__ void lif_step(float x, float& v, float& th, float& s) {
  v  = v + (K_V_REST - v) * K_TAUM_INV + x;           // leak + input
  s  = (v >= th) ? 1.0f : 0.0f;                       // spike
  th = th + K_BETA * s - (th - K_THR0) * K_TAUTH_INV; // threshold adaptation
  v  = (s > 0.0f) ? K_V_REST : v;                     // reset
}

__global__ __launch_bounds__(BLOCK) void lif_scan_kernel(
    const float* __restrict__ in, float* __restrict__ out, int bn) {
  const int tid  = threadIdx.x;
  const int base = (blockIdx.x * BLOCK + tid) * VEC;   // first neuron index
  if (base + VEC > bn) return;

  const float* gsrc = in + base;   // advances by bn floats per timestep
  float*       gdst = out + base;

  // Per-neuron state lives in VGPRs for the whole scan.
  float v0 = K_V_REST, v1 = K_V_REST;
  float th0 = K_THR0,  th1 = K_THR0;

#if ASYNC_PATH
  // DEPTH-deep ring: each lane owns one b64 slot entry per timestep.
  // Lanes only read back the LDS words they themselves fetched, so the
  // wave-private ASYNCcnt is the only synchronization needed (no barriers).
  __shared__ float lds[DEPTH][BLOCK * VEC];   // 16 * 2 KB = 32 KB

  // Prologue: fill the ring DEPTH timesteps deep.
  for (int k = 0; k < DEPTH; ++k) {
    __builtin_amdgcn_global_load_async_to_lds_b64(
        (gv2i_ptr)(uintptr_t)(gsrc + (long)k * bn),   // flat VA == global VA
        (lv2i_ptr)&lds[k][tid * VEC], /*offset=*/0, /*cpol=*/0);
  }

  int t = 0;
  for (; t < T_STEPS - DEPTH; ++t) {
    // Loads complete in order: <= DEPTH-1 outstanding => load t has landed.
    __builtin_amdgcn_s_wait_asynccnt(DEPTH - 1);
    asm volatile("" ::: "memory");

    float2 x = *(const float2*)&lds[t & SLOT_MASK][tid * VEC];

    float s0, s1;
    lif_step(x.x, v0, th0, s0);
    lif_step(x.y, v1, th1, s1);

    float2 sv; sv.x = s0; sv.y = s1;
    *(float2*)(gdst + (long)t * bn) = sv;

    // Async LDS writes are not ordered vs in-flight ds_loads: make sure our
    // read of this slot has retired before the engine overwrites it.
#if defined(__has_builtin) && __has_builtin(__builtin_amdgcn_s_wait_dscnt)
    __builtin_amdgcn_s_wait_dscnt(0);
#else
    asm volatile("s_wait_dscnt 0" ::: "memory");
#endif
    __builtin_amdgcn_global_load_async_to_lds_b64(
        (gv2i_ptr)(uintptr_t)(gsrc + (long)(t + DEPTH) * bn),
        (lv2i_ptr)&lds[t & SLOT_MASK][tid * VEC], 0, 0);
  }

  // Drain: everything for the last DEPTH steps is already in (or heading to)
  // the ring; wait once, then consume without further async traffic.
  __builtin_amdgcn_s_wait_asynccnt(0);
  asm volatile("" ::: "memory");
  for (; t < T_STEPS; ++t) {
    float2 x = *(const float2*)&lds[t & SLOT_MASK][tid * VEC];
    float s0, s1;
    lif_step(x.x, v0, th0, s0);
    lif_step(x.y, v1, th1, s1);
    float2 sv; sv.x = s0; sv.y = s1;
    *(float2*)(gdst + (long)t * bn) = sv;
  }
#else
  // Fallback: plain coalesced loads with gfx1250 global_prefetch_b8 lookahead.
  for (int t = 0; t < T_STEPS; ++t) {
    if (t + 8 < T_STEPS)
      __builtin_prefetch(gsrc + (long)(t + 8) * bn, 0, 0);
    float2 x = *(const float2*)(gsrc + (long)t * bn);
    float s0, s1;
    lif_step(x.x, v0, th0, s0);
    lif_step(x.y, v1, th1, s1);
    float2 sv; sv.x = s0; sv.y = s1;
    *(float2*)(gdst + (long)t * bn) = sv;
  }
#endif
}

extern "C" void kernel_launch(void* const* d_in, const int* in_sizes, int n_in,
                              void* d_out, int out_size, void* d_ws, size_t ws_size,
                              hipStream_t stream) {
  (void)n_in; (void)out_size; (void)d_ws; (void)ws_size;
  const float* in  = (const float*)d_in[0];
  float*       out = (float*)d_out;

  const int total = in_sizes[0];        // T * B * N
  const int bn    = total / T_STEPS;    // B * N = 65536

  const int threads = bn / VEC;                       // 32768
  const int blocks  = (threads + BLOCK - 1) / BLOCK;  // 128

  lif_scan_kernel<<<blocks, BLOCK, 0, stream>>>(in, out, bn);
}